// HardLabel_35433480192258
// MI455X (gfx1250) — compile-verified
//
#include <hip/hip_runtime.h>
#include <stdint.h>
#include <stddef.h>

// Problem constants (from reference): N=8, C=22, H=480, W=640, THRESHOLD=0.9
#define HW_      307200          // H*W
#define C_       22
#define N_       8
#define THR      0.9f
#define DEPTH    4               // async pipeline depth (channels in flight)

typedef float vf4 __attribute__((ext_vector_type(4)));

// One pipeline stage of the label scan:
//  - wait until channel `c`'s async copy has landed in LDS (per-wave, in-order ASYNCcnt)
//  - read this thread's 16B slice (ds_load_b128) and update argmax state
//  - re-issue the freed LDS slot for channel c+DEPTH (after s_wait_dscnt 0 so the
//    LDS read has completed before the TDM-side overwrite)
#define LABEL_STEP(c, wimm)                                                     \
  do {                                                                          \
    asm volatile("s_wait_asynccnt " #wimm ::: "memory");                        \
    vf4 lv = *(const vf4 *)&sbuf[(c) & (DEPTH - 1)][t * 4];                     \
    if (!f0 && lv.x > 0.0f) { gt0 = (c); f0 = true; }                           \
    if (!f1 && lv.y > 0.0f) { gt1 = (c); f1 = true; }                           \
    if (!f2 && lv.z > 0.0f) { gt2 = (c); f2 = true; }                           \
    if (!f3 && lv.w > 0.0f) { gt3 = (c); f3 = true; }                           \
    if ((c) + DEPTH < C_) {                                                     \
      asm volatile("s_wait_dscnt 0" ::: "memory");                              \
      uint32_t lds_a = (uint32_t)(uintptr_t)(&sbuf[(c) & (DEPTH - 1)][t * 4]);  \
      uint64_t ga    = (uint64_t)(uintptr_t)(gsrc0 + (size_t)((c) + DEPTH) * HW_); \
      asm volatile("global_load_async_to_lds_b128 %0, %1, off"                  \
                   :: "v"(lds_a), "v"(ga) : "memory");                          \
    }                                                                           \
  } while (0)

__global__ __launch_bounds__(256) void hardlabel_kernel(
    const float *__restrict__ prob,
    const float *__restrict__ label,
    const float *__restrict__ rnd,
    float *__restrict__ out)
{
  // Quad-buffered (DEPTH-deep ring) channel slabs: DEPTH x 1024 pixels x 4B = 16 KB LDS.
  __shared__ float sbuf[DEPTH][1024];

  const int t = threadIdx.x;            // 0..255, owns one float4 quad of pixels
  const int b = blockIdx.x;             // 2400 blocks; 300 blocks per (n) plane
  const int n = b / 300;                // image index
  const int r = (b % 300) * 1024 + t * 4; // in-plane pixel offset of this quad
  const size_t base0 = (size_t)n * (C_ * HW_) + r; // element index at channel 0

  const float *gsrc0 = label + base0;

  // Prologue: kick DEPTH async channel copies into LDS (HBM -> LDS, no VGPRs held).
#pragma unroll
  for (int c = 0; c < DEPTH; ++c) {
    uint32_t lds_a = (uint32_t)(uintptr_t)(&sbuf[c][t * 4]);
    uint64_t ga    = (uint64_t)(uintptr_t)(gsrc0 + (size_t)c * HW_);
    asm volatile("global_load_async_to_lds_b128 %0, %1, off"
                 :: "v"(lds_a), "v"(ga) : "memory");
  }

  // Overlap the per-pixel rand load (streaming, NT hint) with the async copies.
  vf4 rv = __builtin_nontemporal_load((const vf4 *)(rnd + (size_t)n * HW_ + r));

  int  gt0 = 0, gt1 = 0, gt2 = 0, gt3 = 0;
  bool f0 = false, f1 = false, f2 = false, f3 = false;

  // 22-channel scan, 4 channels in flight. Waits taper at the tail so the
  // in-order ASYNCcnt actually guarantees channel c has landed.
  LABEL_STEP(0, 0x3);  LABEL_STEP(1, 0x3);  LABEL_STEP(2, 0x3);  LABEL_STEP(3, 0x3);
  LABEL_STEP(4, 0x3);  LABEL_STEP(5, 0x3);  LABEL_STEP(6, 0x3);  LABEL_STEP(7, 0x3);
  LABEL_STEP(8, 0x3);  LABEL_STEP(9, 0x3);  LABEL_STEP(10, 0x3); LABEL_STEP(11, 0x3);
  LABEL_STEP(12, 0x3); LABEL_STEP(13, 0x3); LABEL_STEP(14, 0x3); LABEL_STEP(15, 0x3);
  LABEL_STEP(16, 0x3); LABEL_STEP(17, 0x3); LABEL_STEP(18, 0x3);
  LABEL_STEP(19, 0x2); LABEL_STEP(20, 0x1); LABEL_STEP(21, 0x0);

  // Short-circuit: only gather prob[gt] where rand >= THR (≈10% of pixels).
  // Saves ~190 MB of HBM traffic vs reading the full prob tensor.
  const bool e0 = rv.x < THR, e1 = rv.y < THR, e2 = rv.z < THR, e3 = rv.w < THR;
  float pg0 = 1.0f, pg1 = 1.0f, pg2 = 1.0f, pg3 = 1.0f; // >= THR => no effect
  if (f0 && !e0) pg0 = __builtin_nontemporal_load(prob + base0 + (size_t)gt0 * HW_ + 0);
  if (f1 && !e1) pg1 = __builtin_nontemporal_load(prob + base0 + (size_t)gt1 * HW_ + 1);
  if (f2 && !e2) pg2 = __builtin_nontemporal_load(prob + base0 + (size_t)gt2 * HW_ + 2);
  if (f3 && !e3) pg3 = __builtin_nontemporal_load(prob + base0 + (size_t)gt3 * HW_ + 3);

  const float o0 = (f0 && (e0 || pg0 < THR)) ? 1.0f : 0.0f;
  const float o1 = (f1 && (e1 || pg1 < THR)) ? 1.0f : 0.0f;
  const float o2 = (f2 && (e2 || pg2 < THR)) ? 1.0f : 0.0f;
  const float o3 = (f3 && (e3 || pg3 < THR)) ? 1.0f : 0.0f;

  // One-hot write-out: 22 coalesced NT b128 stores per thread (216 MB total).
  float *ob = out + base0;
#pragma unroll
  for (int c = 0; c < C_; ++c) {
    vf4 ov;
    ov.x = (c == gt0) ? o0 : 0.0f;
    ov.y = (c == gt1) ? o1 : 0.0f;
    ov.z = (c == gt2) ? o2 : 0.0f;
    ov.w = (c == gt3) ? o3 : 0.0f;
    __builtin_nontemporal_store(ov, (vf4 *)(ob + (size_t)c * HW_));
  }
}

extern "C" void kernel_launch(void *const *d_in, const int *in_sizes, int n_in,
                              void *d_out, int out_size, void *d_ws, size_t ws_size,
                              hipStream_t stream)
{
  const float *prob  = (const float *)d_in[0];
  const float *label = (const float *)d_in[1];
  const float *rnd   = (const float *)d_in[2];
  float *out = (float *)d_out;

  // N*HW / (256 threads * 4 pixels) = 8*307200/1024 = 2400 blocks.
  dim3 grid(N_ * (HW_ / 1024));
  dim3 block(256);
  hipLaunchKernelGGL(hardlabel_kernel, grid, block, 0, stream, prob, label, rnd, out);
}